// Model_27041114096025
// MI455X (gfx1250) — compile-verified
//
#include <hip/hip_runtime.h>
#include <stdint.h>

#define NDIM 1024
#define TILE 128
#define BK   32
#define ASTRIDE 36     // 128x32 A tile, padded row stride (even, conflict-free, 16B aligned)
#define BSTRIDE 288    // pair-interleaved B tile rows: 2*128 + 32 (disjoint banks for lane halves)

typedef __attribute__((ext_vector_type(2))) float v2f;
typedef __attribute__((ext_vector_type(4))) float v4f;
typedef __attribute__((ext_vector_type(8))) float v8f;
typedef int vi4 __attribute__((__vector_size__(16)));   // matches builtin param type

#if defined(__gfx1250__) && __has_builtin(__builtin_amdgcn_global_load_async_to_lds_b128)
#define HAVE_ASYNC_LDS 1
#else
#define HAVE_ASYNC_LDS 0
#endif

#if HAVE_ASYNC_LDS
typedef __attribute__((address_space(1))) vi4 gvi4;
typedef __attribute__((address_space(3))) vi4 lvi4;
__device__ __forceinline__ void async_copy_b128(const float* src, void* lds_dst) {
    // AS1 pointer: global generic addresses are identical to AS1.
    // AS3 pointer: LDS aperture address is addr[31:0] (ISA 10.2), i.e. low dword.
    __builtin_amdgcn_global_load_async_to_lds_b128(
        (gvi4*)(uintptr_t)src,
        (lvi4*)(uint32_t)(uintptr_t)lds_dst,
        /*offset=*/0, /*cpol=*/0);
}
__device__ __forceinline__ void wait_async0() {
#if __has_builtin(__builtin_amdgcn_s_wait_asynccnt)
    __builtin_amdgcn_s_wait_asynccnt(0);
#else
    asm volatile("s_wait_asynccnt 0x0" ::: "memory");
#endif
}
#endif

struct GemmArgs {
    const float* X[7];
    const float* Y[7];
    float*       O[7];
};

// ---------------------------------------------------------------------------
// Batched FP32 GEMM: O[p] = X[p] @ Y[p], p = blockIdx.z in [0,7)
// 256 threads = 8 waves; block tile 128x128; K-block 32; double-buffered LDS.
// A tile: GLOBAL_LOAD_ASYNC_TO_LDS_B128 (ASYNCcnt) when available.
// B tile: register-staged, stored K-pair-interleaved -> single ds_load_b64 frags.
// Each wave: 2x4 grid of 16x16 accumulators -> v_wmma_f32_16x16x4_f32.
// ---------------------------------------------------------------------------
__launch_bounds__(256, 2)
__global__ void gemm7_f32_wmma(GemmArgs args) {
    __shared__ float As[2][TILE][ASTRIDE];       // 2 x 18 KB
    __shared__ float Bs[2][BK / 2][BSTRIDE];     // 2 x 18 KB  (pair-interleaved)

    const int p = blockIdx.z;
    const float* __restrict__ X = args.X[p];
    const float* __restrict__ Y = args.Y[p];
    float*       __restrict__ O = args.O[p];

    const int t       = threadIdx.x;
    const int lane    = t & 31;
    const int wave    = t >> 5;
    const int waveRow = wave >> 1;        // 0..3 -> 32 output rows each
    const int waveCol = wave & 1;         // 0..1 -> 64 output cols each
    const int tileRow = blockIdx.y * TILE;
    const int tileCol = blockIdx.x * TILE;

    const int lhalf = lane >> 4;          // 0 | 1 : selects K-pair (ISA A/B layout)
    const int lmod  = lane & 15;          // M (A frag) / N (B frag)

    v8f acc[2][4];
    #pragma unroll
    for (int i = 0; i < 2; ++i)
        #pragma unroll
        for (int j = 0; j < 4; ++j)
            acc[i][j] = {};

    v4f rb[4];                            // register staging for B double buffer
#if !HAVE_ASYNC_LDS
    v4f ra[4];                            // fallback: register staging for A too
#endif

    // ---- A tile 128x32 -> LDS buffer `buf` (async direct, or via regs) ----
    auto loadA = [&](int kb, int buf) {
        #pragma unroll
        for (int i = 0; i < 4; ++i) {
            int idx = t + 256 * i;                    // 0..1023
            int rA  = idx >> 3;                       // 0..127
            int cA  = (idx & 7) * 4;                  // 0..28
            const float* src = X + (size_t)(tileRow + rA) * NDIM + kb + cA;
#if HAVE_ASYNC_LDS
            async_copy_b128(src, &As[buf][rA][cA]);   // global -> LDS, no VGPRs
#else
            ra[i] = *(const v4f*)src;
            (void)buf;
#endif
        }
    };
#if !HAVE_ASYNC_LDS
    auto spillA = [&](int buf) {
        #pragma unroll
        for (int i = 0; i < 4; ++i) {
            int idx = t + 256 * i;
            int rA  = idx >> 3;
            int cA  = (idx & 7) * 4;
            *(v4f*)&As[buf][rA][cA] = ra[i];
        }
    };
#endif
    // ---- B tile 32x128 -> regs ----
    auto loadBRegs = [&](int kb) {
        #pragma unroll
        for (int i = 0; i < 4; ++i) {
            int idx = t + 256 * i;
            int rB  = idx >> 5;                       // 0..31
            int cB  = (idx & 31) * 4;                 // 0..124
            rb[i] = *(const v4f*)(Y + (size_t)(kb + rB) * NDIM + tileCol + cB);
        }
    };
    // ---- regs -> pair-interleaved LDS buffer `buf` ----
    auto spillB = [&](int buf) {
        #pragma unroll
        for (int i = 0; i < 4; ++i) {
            int idx = t + 256 * i;
            int rB  = idx >> 5;
            int cB  = (idx & 31) * 4;
            int kp  = rB >> 1;                        // K-pair row
            int par = rB & 1;                         // position within pair
            #pragma unroll
            for (int j = 0; j < 4; ++j)
                Bs[buf][kp][2 * (cB + j) + par] = rb[i][j];
        }
    };

    // prologue: fill buffer 0
    loadA(0, 0);
    loadBRegs(0);
    spillB(0);
#if !HAVE_ASYNC_LDS
    spillA(0);
#else
    wait_async0();
#endif
    __syncthreads();

    int cur = 0;
    for (int kb = 0; kb < NDIM; kb += BK) {
        const bool hasNext = (kb + BK) < NDIM;
        if (hasNext) {
            loadA(kb + BK, 1 - cur);      // async A loads land during compute
            loadBRegs(kb + BK);           // B global loads in flight during compute
        }

        #pragma unroll
        for (int kk = 0; kk < BK; kk += 4) {
            const int k0 = kk + lhalf * 2;            // even -> 8B-aligned pair
            const int kp = (kk >> 1) + lhalf;         // B pair row
            v2f a[2], b[4];
            #pragma unroll
            for (int i = 0; i < 2; ++i) {
                int m = waveRow * 32 + i * 16 + lmod;
                a[i] = *(const v2f*)&As[cur][m][k0];
            }
            #pragma unroll
            for (int j = 0; j < 4; ++j) {
                int n = waveCol * 64 + j * 16 + lmod;
                b[j] = *(const v2f*)&Bs[cur][kp][2 * n];
            }
            #pragma unroll
            for (int i = 0; i < 2; ++i)
                #pragma unroll
                for (int j = 0; j < 4; ++j)
                    acc[i][j] = __builtin_amdgcn_wmma_f32_16x16x4_f32(
                        /*neg_a=*/false, a[i], /*neg_b=*/false, b[j],
                        /*c_mod=*/(short)0, acc[i][j],
                        /*reuse_a=*/false, /*reuse_b=*/false);
        }

        if (hasNext) {
            spillB(1 - cur);              // other buffer: no wait on readers
#if !HAVE_ASYNC_LDS
            spillA(1 - cur);
#else
            wait_async0();                // drain ASYNCcnt before readers flip
#endif
        }
        __syncthreads();
        cur ^= 1;
    }

    // ---- store: C/D layout — vgpr r holds row r (lanes 0-15) / r+8 (lanes 16-31) ----
    #pragma unroll
    for (int i = 0; i < 2; ++i) {
        #pragma unroll
        for (int j = 0; j < 4; ++j) {
            const int col = tileCol + waveCol * 64 + j * 16 + lmod;
            #pragma unroll
            for (int r = 0; r < 8; ++r) {
                const int row = tileRow + waveRow * 32 + i * 16 + r + lhalf * 8;
                O[(size_t)row * NDIM + col] = acc[i][j][r];
            }
        }
    }
}

// ---------------------------------------------------------------------------
// Elementwise epilogue: t137 from A..G (float4 vectorized).
// ---------------------------------------------------------------------------
__launch_bounds__(256)
__global__ void combine_t137(const float* __restrict__ Ap, const float* __restrict__ Bp,
                             const float* __restrict__ Cp, const float* __restrict__ Dp,
                             const float* __restrict__ Ep, const float* __restrict__ Fp,
                             const float* __restrict__ Gp, float* __restrict__ out) {
    const size_t i4 = (size_t)blockIdx.x * blockDim.x + threadIdx.x;
    const size_t off = i4 * 4;

    v4f a = *(const v4f*)(Ap + off);
    v4f b = *(const v4f*)(Bp + off);
    v4f c = *(const v4f*)(Cp + off);
    v4f d = *(const v4f*)(Dp + off);
    v4f e = *(const v4f*)(Ep + off);
    v4f f = *(const v4f*)(Fp + off);
    v4f g = *(const v4f*)(Gp + off);

    v4f t37  = a + 2.0f * b;
    v4f t65  = a + c;
    v4f t66  = t37 + t65;
    v4f t68  = t66 * (-b);
    v4f t71  = d + g;
    v4f t74  = t68 + t71;
    v4f t75  = b - t71;
    v4f t76  = t68 + (a + e + 2.0f * f);
    v4f t77  = t75 + t76;
    v4f t79  = e + f;
    v4f t81  = t65 * t79;
    v4f t82  = t77 + (a + t68);
    v4f t83  = t79 + t82;
    v4f t84  = t81 + t83;
    v4f t85  = t84 * t74;
    v4f t86  = t85 - g;
    v4f t87  = t85 + g;
    v4f t88  = t84 * t77;
    v4f t89  = b - t79;
    v4f t90  = t85 * t76;
    v4f t92  = t90 + (t88 + t89);
    v4f t93  = t92 * t74;
    v4f t94  = t93 * f;
    v4f t95  = t93 * e;
    v4f t97  = t81 * a;
    v4f t101 = t87 + ((t93 - t94) + ((t95 - e) + t97));
    v4f t102 = t87 * t86;
    v4f t103 = t101 * t79;
    v4f t104 = t102 * e;
    v4f t105 = t103 * e;
    v4f t106 = (t104 - t105) * (t89 * e) + (t103 + t105) * (t90 * f);
    v4f t107 = t106;
    v4f t108 = t107 * t88;
    v4f t109 = t92 * t88;
    v4f t110 = t107 * t95;
    v4f t111 = t97 * t92;
    v4f t116 = (t108 - t109) * t86;
    v4f t117 = (t110 - t111) * t86;
    v4f t118 = t116 + t117;
    v4f t119 = t107 * t101;
    v4f t120 = t118 + t119;
    v4f t121 = t107 + t120;
    v4f t124 = (t119 * t88) * t86;
    v4f t125 = (t118 * t95) * t86;
    v4f t130 = (t107 * t88) * t86 - (t107 * t95) * t86;
    v4f t132 = t130 + (t124 + t125);
    v4f t133 = t107 - t121;
    v4f t134 = t132 - t133;
    v4f t137 = t134 * t86 - t134 * t97;

    *(v4f*)(out + off) = t137;
}

extern "C" void kernel_launch(void* const* d_in, const int* in_sizes, int n_in,
                              void* d_out, int out_size, void* d_ws, size_t ws_size,
                              hipStream_t stream) {
    const float* i1 = (const float*)d_in[0];
    const float* i2 = (const float*)d_in[1];
    const float* i3 = (const float*)d_in[2];

    const size_t M = (size_t)NDIM * NDIM;
    float* ws = (float*)d_ws;
    float* A = ws + 0 * M;   // i1 @ i2
    float* B = ws + 1 * M;   // i1 @ i3
    float* C = ws + 2 * M;   // i1 @ i1
    float* D = ws + 3 * M;   // i2 @ i1
    float* E = ws + 4 * M;   // i2 @ i2
    float* F = ws + 5 * M;   // i2 @ i3
    float* G = ws + 6 * M;   // i3 @ i2

    GemmArgs ga;
    ga.X[0] = i1; ga.Y[0] = i2; ga.O[0] = A;
    ga.X[1] = i1; ga.Y[1] = i3; ga.O[1] = B;
    ga.X[2] = i1; ga.Y[2] = i1; ga.O[2] = C;
    ga.X[3] = i2; ga.Y[3] = i1; ga.O[3] = D;
    ga.X[4] = i2; ga.Y[4] = i2; ga.O[4] = E;
    ga.X[5] = i2; ga.Y[5] = i3; ga.O[5] = F;
    ga.X[6] = i3; ga.Y[6] = i2; ga.O[6] = G;

    dim3 grid(NDIM / TILE, NDIM / TILE, 7);
    hipLaunchKernelGGL(gemm7_f32_wmma, grid, dim3(256), 0, stream, ga);

    const size_t nvec = M / 4;            // float4 elements
    hipLaunchKernelGGL(combine_t137, dim3((unsigned)(nvec / 256)), dim3(256), 0, stream,
                       A, B, C, D, E, F, G, (float*)d_out);
}